// NGM_Net_18829136625934
// MI455X (gfx1250) — compile-verified
//
#include <hip/hip_runtime.h>

// ---------------------------------------------------------------------------
// NGM forward for MI455X (gfx1250, wave32).
//  - deg:      wave-per-row nnz count of K          (reads K once, 164 MB)
//  - mlp:      per-node 2-layer ReLU MLPs (n & self branches)
//  - spmm:     y = K @ x1 via V_WMMA_F32_16X16X4_F32 with double-buffered
//              LDS staging (global loads for chunk n+1 overlap WMMA of n),
//              row-scale by 1/deg, add self branch, write emb ch [0..15]
//  - score:    per-node dot with cw (16) / fw (17)
//  - sinkhorn: one block per batch, 40x40 in LDS, 20 alternating
//              row/col logsumexp normalizations (log-space, dummy rows -100)
// ---------------------------------------------------------------------------

typedef float v2f __attribute__((ext_vector_type(2)));
typedef float v4f __attribute__((ext_vector_type(4)));
typedef float v8f __attribute__((ext_vector_type(8)));

#define NB    16      // batch
#define NM    40      // n1max == n2max
#define NN    1600    // NM*NM nodes
#define CH    16      // gnn channels
#define EMBC  17      // CH + 1 (x4 channel)
#define KB    64      // k-chunk staged in LDS
#define NCHUNK (NN / KB)
#define APAD  68      // padded LDS row stride for K tile (bank-conflict free)

// ---------------- deg: 1 / max(row nnz, 1e-12) -----------------------------
__global__ __launch_bounds__(256)
void deg_kernel(const float* __restrict__ K, float* __restrict__ deg_inv) {
    int gid  = blockIdx.x * blockDim.x + threadIdx.x;
    int row  = gid >> 5;
    int lane = gid & 31;
    if (row >= NB * NN) return;
    const float* r = K + (size_t)row * NN;
    float cnt = 0.0f;
    for (int j = lane; j < NN; j += 32)
        cnt += (r[j] != 0.0f) ? 1.0f : 0.0f;
    for (int off = 16; off > 0; off >>= 1)
        cnt += __shfl_down(cnt, off, 32);
    if (lane == 0) deg_inv[row] = 1.0f / fmaxf(cnt, 1e-12f);
}

// ---------------- per-node 2-layer MLPs (n-branch and self-branch) ---------
__global__ __launch_bounds__(256)
void mlp_kernel(const float* __restrict__ embp, int in_f, int in_stride,
                const float* __restrict__ w1, const float* __restrict__ b1,
                const float* __restrict__ w2, const float* __restrict__ b2,
                const float* __restrict__ s1, const float* __restrict__ t1,
                const float* __restrict__ s2, const float* __restrict__ t2,
                float* __restrict__ x1o, float* __restrict__ selfo) {
    int node = blockIdx.x * blockDim.x + threadIdx.x;
    if (node >= NB * NN) return;
    float e[EMBC];
    const float* er = embp + (size_t)node * in_stride;
    for (int i = 0; i < in_f; ++i) e[i] = er[i];

    float h[CH];
    // n-branch
    for (int c = 0; c < CH; ++c) {
        float acc = b1[c];
        for (int i = 0; i < in_f; ++i) acc += e[i] * w1[i * CH + c];
        h[c] = fmaxf(acc, 0.0f);
    }
    for (int c = 0; c < CH; ++c) {
        float acc = b2[c];
        for (int i = 0; i < CH; ++i) acc += h[i] * w2[i * CH + c];
        x1o[(size_t)node * CH + c] = fmaxf(acc, 0.0f);
    }
    // self-branch
    for (int c = 0; c < CH; ++c) {
        float acc = t1[c];
        for (int i = 0; i < in_f; ++i) acc += e[i] * s1[i * CH + c];
        h[c] = fmaxf(acc, 0.0f);
    }
    for (int c = 0; c < CH; ++c) {
        float acc = t2[c];
        for (int i = 0; i < CH; ++i) acc += h[i] * s2[i * CH + c];
        selfo[(size_t)node * CH + c] = fmaxf(acc, 0.0f);
    }
}

// ---------------- spmm: emb[:, :, 0:16] = (K @ x1) / deg + self ------------
// 128 threads = 4 waves; each wave owns one 16-row M tile, N = 16 channels
// (one WMMA tile). Double-buffered LDS: global loads for the next 64-wide
// K chunk are issued before computing the current chunk.
__global__ __launch_bounds__(128)
void spmm_wmma_kernel(const float* __restrict__ K, const float* __restrict__ x1,
                      const float* __restrict__ deg_inv,
                      const float* __restrict__ selfT,
                      float* __restrict__ emb) {
    __shared__ float ldsB[2][KB * CH];        // x1 chunk: 64 x 16
    __shared__ float ldsA[2][4][16 * APAD];   // per-wave K tile: 16 x 64 (padded)

    int b    = blockIdx.x / (NN / 64);
    int mb   = blockIdx.x % (NN / 64);
    int t    = threadIdx.x;
    int wave = t >> 5, lane = t & 31;
    int m_base = mb * 64 + wave * 16;

    const float* Kb  = K  + (size_t)b * NN * NN;
    const float* x1b = x1 + (size_t)b * NN * CH;

    // staging addresses (fixed per thread)
    int kr  = lane >> 1;                      // K tile row 0..15
    int kc0 = (lane & 1) * 32;                // K tile col half
    const float* ksrc0 = Kb + (size_t)(m_base + kr) * NN + kc0;
    const float* xsrc0 = x1b + t * 8;         // 8 consecutive floats / thread

    v4f ka[8];                                // 32 floats of K per thread
    v4f xb[2];                                // 8 floats of x1 per thread

#define LOAD_REGS(kk)                                              \
    {                                                              \
        const float* ks = ksrc0 + (kk);                            \
        for (int j = 0; j < 8; ++j)                                \
            ka[j] = *(const v4f*)(ks + j * 4);                     \
        const float* xs = xsrc0 + (size_t)(kk) * CH;               \
        xb[0] = *(const v4f*)(xs);                                 \
        xb[1] = *(const v4f*)(xs + 4);                             \
    }

#define STORE_LDS(buf)                                             \
    {                                                              \
        float* dst = &ldsA[buf][wave][kr * APAD + kc0];            \
        for (int j = 0; j < 8; ++j)                                \
            *(v4f*)(dst + j * 4) = ka[j];                          \
        *(v4f*)&ldsB[buf][t * 8]     = xb[0];                      \
        *(v4f*)&ldsB[buf][t * 8 + 4] = xb[1];                      \
    }

    v8f acc = {};
    int ml    = lane & 15;           // M (A frag) / N (B frag) index
    int khalf = (lane >> 4) * 2;     // k sub-offset: lanes 0-15 -> 0, 16-31 -> 2

    LOAD_REGS(0);
    STORE_LDS(0);
    __syncthreads();

    for (int n = 0; n < NCHUNK; ++n) {
        int  cur  = n & 1;
        bool more = (n + 1) < NCHUNK;
        if (more) LOAD_REGS((n + 1) * KB);    // overlap with WMMA below

        for (int k4 = 0; k4 < KB; k4 += 4) {
            v2f a, bb;
            a.x  = ldsA[cur][wave][ml * APAD + k4 + khalf];
            a.y  = ldsA[cur][wave][ml * APAD + k4 + khalf + 1];
            bb.x = ldsB[cur][(k4 + khalf) * CH + ml];
            bb.y = ldsB[cur][(k4 + khalf + 1) * CH + ml];
            acc = __builtin_amdgcn_wmma_f32_16x16x4_f32(
                false, a, false, bb, (short)0, acc, false, false);
        }
        if (more) STORE_LDS(cur ^ 1);
        __syncthreads();
    }
#undef LOAD_REGS
#undef STORE_LDS

    // epilogue: C layout -> VGPR r holds M = r (lanes 0-15) / M = 8+r (16-31)
    int mhi = (lane >> 4) * 8;
    for (int r = 0; r < 8; ++r) {
        int m = m_base + mhi + r;
        float dinv = deg_inv[b * NN + m];
        float v = acc[r] * dinv + selfT[((size_t)b * NN + m) * CH + ml];
        emb[((size_t)b * NN + m) * EMBC + ml] = v;
    }
}

// ---------------- per-node score: sc = bias + emb[0:nc] . w ----------------
__global__ __launch_bounds__(256)
void score_kernel(const float* __restrict__ emb, const float* __restrict__ w,
                  const float* __restrict__ bias, int nc,
                  float* __restrict__ sc) {
    int node = blockIdx.x * blockDim.x + threadIdx.x;
    if (node >= NB * NN) return;
    const float* e = emb + (size_t)node * EMBC;
    float a = bias[0];
    for (int c = 0; c < nc; ++c) a += e[c] * w[c];
    sc[node] = a;
}

// ---------------- sinkhorn: one block (64 thr) per batch -------------------
// s[i][j] = sc[b, j*40 + i] / tau ; rows >= n1 set to -100 ; 20 alternating
// row/col logsumexp subtractions ; output P[i][j] = exp(s) masked.
__global__ __launch_bounds__(64)
void sinkhorn_kernel(const float* __restrict__ sc, const int* __restrict__ n1,
                     const int* __restrict__ maxit_p,
                     const float* __restrict__ tau_p,
                     float* __restrict__ dst, int final_mode) {
    __shared__ float s[NM][NM + 1];
    int b = blockIdx.x;
    int t = threadIdx.x;
    float inv_tau = 1.0f / tau_p[0];
    int maxit = maxit_p[0];
    int nr = n1[b];

    for (int q = t; q < NN; q += 64) {
        int i = q % NM, j = q / NM;
        float v = sc[b * NN + q] * inv_tau;
        if (i >= nr) v = -100.0f;
        s[i][j] = v;
    }
    __syncthreads();

    for (int it = 0; it < maxit; ++it) {
        if (t < NM) {
            if ((it & 1) == 0) {                  // normalize rows (axis 2)
                float m = -3.4e38f;
                for (int j = 0; j < NM; ++j) m = fmaxf(m, s[t][j]);
                float sum = 0.0f;
                for (int j = 0; j < NM; ++j) sum += __expf(s[t][j] - m);
                float lse = m + __logf(sum);
                for (int j = 0; j < NM; ++j) s[t][j] -= lse;
            } else {                               // normalize cols (axis 1)
                float m = -3.4e38f;
                for (int i = 0; i < NM; ++i) m = fmaxf(m, s[i][t]);
                float sum = 0.0f;
                for (int i = 0; i < NM; ++i) sum += __expf(s[i][t] - m);
                float lse = m + __logf(sum);
                for (int i = 0; i < NM; ++i) s[i][t] -= lse;
            }
        }
        __syncthreads();
    }

    for (int q = t; q < NN; q += 64) {
        int i = q % NM, j = q / NM;
        float p = (i < nr) ? __expf(s[i][j]) : 0.0f;   // cols always valid (n2==40)
        if (final_mode) dst[b * NN + i * NM + j] = p;             // (B, 40, 40)
        else            dst[((size_t)b * NN + q) * EMBC + CH] = p; // emb ch16, idx=j*40+i
    }
}

// ---------------------------------------------------------------------------
extern "C" void kernel_launch(void* const* d_in, const int* in_sizes, int n_in,
                              void* d_out, int out_size, void* d_ws, size_t ws_size,
                              hipStream_t stream) {
    // setup_inputs() order: K, n1, n2, n1max, n2max, v0, sk_max_iter, sk_tau,
    // then params pytree leaves (dict keys sorted): fb, fw,
    // layers[0..2] each: cb, cw, w1, b1, w2, b2, s1, t1, s2, t2
    const float* K     = (const float*)d_in[0];
    const int*   n1    = (const int*)d_in[1];
    const float* v0    = (const float*)d_in[5];
    const int*   maxit = (const int*)d_in[6];
    const float* tau   = (const float*)d_in[7];
    const float* fb    = (const float*)d_in[8];
    const float* fw    = (const float*)d_in[9];

    float* ws      = (float*)d_ws;
    float* deg_inv = ws;                                  // NB*NN
    float* emb     = deg_inv + NB * NN;                   // NB*NN*17
    float* x1      = emb + (size_t)NB * NN * EMBC;        // NB*NN*16
    float* selfT   = x1  + (size_t)NB * NN * CH;          // NB*NN*16
    float* sc      = selfT + (size_t)NB * NN * CH;        // NB*NN

    deg_kernel<<<(NB * NN * 32) / 256, 256, 0, stream>>>(K, deg_inv);

    const float* embp = v0;
    int in_f = 1, estride = 1;
    for (int L = 0; L < 3; ++L) {
        int p = 10 + L * 10;
        const float* cb = (const float*)d_in[p + 0];
        const float* cw = (const float*)d_in[p + 1];
        const float* w1 = (const float*)d_in[p + 2];
        const float* b1 = (const float*)d_in[p + 3];
        const float* w2 = (const float*)d_in[p + 4];
        const float* b2 = (const float*)d_in[p + 5];
        const float* s1 = (const float*)d_in[p + 6];
        const float* t1 = (const float*)d_in[p + 7];
        const float* s2 = (const float*)d_in[p + 8];
        const float* t2 = (const float*)d_in[p + 9];

        mlp_kernel<<<(NB * NN + 255) / 256, 256, 0, stream>>>(
            embp, in_f, estride, w1, b1, w2, b2, s1, t1, s2, t2, x1, selfT);
        spmm_wmma_kernel<<<NB * (NN / 64), 128, 0, stream>>>(
            K, x1, deg_inv, selfT, emb);
        score_kernel<<<(NB * NN + 255) / 256, 256, 0, stream>>>(
            emb, cw, cb, CH, sc);
        sinkhorn_kernel<<<NB, 64, 0, stream>>>(
            sc, n1, maxit, tau, emb, 0);

        embp = emb; in_f = EMBC; estride = EMBC;
    }

    score_kernel<<<(NB * NN + 255) / 256, 256, 0, stream>>>(
        emb, fw, fb, EMBC, sc);
    sinkhorn_kernel<<<NB, 64, 0, stream>>>(
        sc, n1, maxit, tau, (float*)d_out, 1);
}